// LocalWindowTokenMerging_80272938762628
// MI455X (gfx1250) — compile-verified
//
#include <hip/hip_runtime.h>
#include <hip/hip_bf16.h>
#include <math.h>

// ---- fixed problem geometry (from setup_inputs) ----
#define BATCH   2
#define NTOK    8192
#define DIM     768
#define CDIM    192          // D/4
#define WSZ     16
#define NWINB   (NTOK / WSZ)         // 512 windows per batch
#define NWIN    (BATCH * NWINB)      // 1024 windows total
#define RPW     4
#define NEWW    (WSZ - RPW)          // 12 output rows per window
#define XOUT_ELEMS ((size_t)NWIN * NEWW * DIM)          // 9,437,184
#define COLS_OUT   (NWINB * NEWW)                        // 6144 merged cols / batch

typedef __attribute__((ext_vector_type(2))) float v2f;
typedef __attribute__((ext_vector_type(8))) float v8f;

// -------------------------------------------------------------------------
// Kernel 1: per-window merge plan.  One wave (32 lanes) per window.
//   metric = x_win(16x768) @ W_group^T(768x192) via V_WMMA_F32_16X16X4_F32
//   then normalize, bipartite scores, argmax, stable argsort, emit 12 masks.
// -------------------------------------------------------------------------
__global__ __launch_bounds__(32) void plan_kernel(
    const float* __restrict__ x, const int* __restrict__ amask,
    const float* __restrict__ Wg, unsigned* __restrict__ plan)
{
    const int win  = blockIdx.x;          // 0..NWIN-1
    const int lane = threadIdx.x;         // 0..31
    const int hf   = lane >> 4;           // 0: K=0,1   1: K=2,3
    const int l16  = lane & 15;           // M (for A) / N (for B)

    __shared__ float metric[WSZ * CDIM];  // 16 x 192, 12 KB
    __shared__ float invn[WSZ];
    __shared__ float scores[8 * 8];
    __shared__ float nmax[8];
    __shared__ int   nidx[8];

    const float* xw   = x + (size_t)win * WSZ * DIM;
    const float* arow = xw + (size_t)l16 * DIM;          // A row M = l16

    // 12 column tiles of 16x16
    for (int c = 0; c < CDIM / 16; ++c) {
        const float* brow = Wg + (size_t)(c * 16 + l16) * DIM;  // B col N = l16
        v8f acc = {};
        for (int k = 0; k < DIM; k += 4) {
            // A 16x4 f32: lanes 0-15 K=0,1 ; lanes 16-31 K=2,3 (M = l16)
            v2f a = *(const v2f*)(arow + k + 2 * hf);
            // B 4x16 f32 (mirrored): lanes 0-15 K=0,1 ; lanes 16-31 K=2,3 (N = l16)
            v2f b = *(const v2f*)(brow + k + 2 * hf);
            acc = __builtin_amdgcn_wmma_f32_16x16x4_f32(
                false, a, false, b, (short)0, acc, false, false);
        }
        // C/D layout: VGPR r, lane L -> M = r + 8*(L>>4), N = L&15
        #pragma unroll
        for (int r = 0; r < 8; ++r) {
            int M = r + 8 * hf;
            metric[M * CDIM + c * 16 + l16] = acc[r];
        }
    }
    __syncthreads();

    // row norms (lanes 0..15, one row each)
    if (lane < WSZ) {
        float s = 0.f;
        for (int c = 0; c < CDIM; ++c) { float v = metric[lane * CDIM + c]; s += v * v; }
        invn[lane] = 1.0f / fmaxf(sqrtf(s), 1e-12f);
    }
    __syncthreads();
    for (int i = lane; i < WSZ * CDIM; i += 32) metric[i] *= invn[i / CDIM];
    __syncthreads();

    // 8x8 bipartite scores (even rows vs odd rows) + attention mask
    const int* mw = amask + (size_t)win * WSZ;
    for (int e = lane; e < 64; e += 32) {
        int i = e >> 3, j = e & 7;
        float s = 0.f;
        const float* ra = metric + (2 * i) * CDIM;
        const float* rb = metric + (2 * j + 1) * CDIM;
        for (int c = 0; c < CDIM; ++c) s += ra[c] * rb[c];
        bool valid = (mw[2 * i] != 0) && (mw[2 * j + 1] != 0);
        scores[e] = valid ? s : -__builtin_inff();
    }
    __syncthreads();

    // per-even-row max / argmax (first max, like jnp.argmax)
    if (lane < 8) {
        float m = scores[lane * 8]; int mi = 0;
        for (int j = 1; j < 8; ++j) {
            float v = scores[lane * 8 + j];
            if (v > m) { m = v; mi = j; }
        }
        nmax[lane] = m; nidx[lane] = mi;
    }
    __syncthreads();

    if (lane == 0) {
        // stable descending insertion sort of nmax (ties keep lower index first)
        int ord[8];
        for (int i = 0; i < 8; ++i) ord[i] = i;
        for (int i = 1; i < 8; ++i) {
            int key = ord[i]; float kv = nmax[key]; int j = i - 1;
            while (j >= 0 && nmax[ord[j]] < kv) { ord[j + 1] = ord[j]; --j; }
            ord[j + 1] = key;
        }
        // output rows 0..3  : unmerged even rows (sorted order positions 4..7)
        // output rows 4..11 : odd rows j, plus merged-in even rows
        unsigned masks[NEWW];
        #pragma unroll
        for (int k = 0; k < RPW; ++k) masks[k] = 1u << (2 * ord[RPW + k]);
        #pragma unroll
        for (int j = 0; j < 8; ++j)   masks[RPW + j] = 1u << (2 * j + 1);
        #pragma unroll
        for (int m = 0; m < RPW; ++m) {
            int si = ord[m];            // merged even row
            int dj = nidx[si];          // destination odd row
            masks[RPW + dj] |= 1u << (2 * si);
        }
        #pragma unroll
        for (int k = 0; k < NEWW; ++k) plan[win * NEWW + k] = masks[k];
    }
}

// -------------------------------------------------------------------------
// Kernel 2: merged x.  One block per output row (win*12+k); mean over mask.
// -------------------------------------------------------------------------
__global__ __launch_bounds__(256) void xmerge_kernel(
    const float* __restrict__ x, const unsigned* __restrict__ plan,
    float* __restrict__ out)
{
    const int row = blockIdx.x;               // 0 .. NWIN*12-1
    const int win = row / NEWW;
    const unsigned m = plan[row];
    const float inv = 1.0f / (float)__popc(m);
    const float* xw  = x + (size_t)win * WSZ * DIM;
    float* orow = out + (size_t)row * DIM;
    for (int d = threadIdx.x; d < DIM; d += 256) {
        float s = 0.f;
        unsigned mm = m;
        while (mm) {
            int t = __ffs(mm) - 1; mm &= mm - 1;
            s += xw[t * DIM + d];
        }
        orow[d] = s * inv;
    }
}

// -------------------------------------------------------------------------
// Kernel 3: merged source (amax along token axis) — the bandwidth kernel.
// Each block: one source row (b,s), one chunk of 128 windows.
//   stage 2048 input floats (b128, coalesced) + 1536 plan masks in LDS,
//   then 256 threads emit 1536 coalesced outputs (max over mask bits).
// -------------------------------------------------------------------------
#define WPC       128                  // windows per chunk
#define CHUNK_IN  (WPC * WSZ)          // 2048 floats
#define CHUNK_OUT (WPC * NEWW)         // 1536 floats
#define NCHUNK    (NWINB / WPC)        // 4 chunks per batch-row

__global__ __launch_bounds__(256) void smerge_kernel(
    const float* __restrict__ src, const unsigned* __restrict__ plan,
    float* __restrict__ out)
{
    const int bid   = blockIdx.x;
    const int chunk = bid & (NCHUNK - 1);
    const int row   = bid >> 2;              // b*NTOK + s   (0..16383)
    const int b     = row >> 13;             // row / 8192
    const int tid   = threadIdx.x;

    const float* srow = src + (size_t)row * NTOK + chunk * CHUNK_IN;
    float*       orow = out + (size_t)row * COLS_OUT + chunk * CHUNK_OUT;
    const int win0 = b * NWINB + chunk * WPC;       // first global window in chunk

    __shared__ float    sin[CHUNK_IN];      // 8 KB
    __shared__ unsigned spl[CHUNK_OUT];     // 6 KB

    const float4* s4 = (const float4*)srow;
    float4*       l4 = (float4*)sin;
    for (int i = tid; i < CHUNK_IN / 4; i += 256) l4[i] = s4[i];
    for (int i = tid; i < CHUNK_OUT; i += 256)    spl[i] = plan[(size_t)win0 * NEWW + i];
    __syncthreads();

    for (int c = tid; c < CHUNK_OUT; c += 256) {
        const int wloc = c / NEWW;
        unsigned m = spl[c];
        const float* base = sin + wloc * WSZ;
        float v = -__builtin_inff();
        while (m) {
            int t = __ffs(m) - 1; m &= m - 1;
            v = fmaxf(v, base[t]);
        }
        orow[c] = v;
    }
}

// -------------------------------------------------------------------------
extern "C" void kernel_launch(void* const* d_in, const int* in_sizes, int n_in,
                              void* d_out, int out_size, void* d_ws, size_t ws_size,
                              hipStream_t stream)
{
    const float* x      = (const float*)d_in[0];
    const float* source = (const float*)d_in[1];
    const int*   amask  = (const int*)d_in[2];
    const float* Wg     = (const float*)d_in[3];
    // d_in[4] = r (always 4 for this problem; r_pw = min(4, 8) = 4)

    unsigned* plan = (unsigned*)d_ws;              // NWIN*12 uints = 48 KB
    float* xout = (float*)d_out;                   // (B, nw*12, D)
    float* sout = (float*)d_out + XOUT_ELEMS;      // (B, N, nw*12)

    plan_kernel  <<<NWIN, 32, 0, stream>>>(x, amask, Wg, plan);
    xmerge_kernel<<<NWIN * NEWW, 256, 0, stream>>>(x, plan, xout);
    smerge_kernel<<<BATCH * NTOK * NCHUNK, 256, 0, stream>>>(source, plan, sout);
}